// Classifier_GNN_16381005267220
// MI455X (gfx1250) — compile-verified
//
#include <hip/hip_runtime.h>
#include <hip/hip_bf16.h>
#include <math.h>

typedef __attribute__((ext_vector_type(2))) float v2f;
typedef __attribute__((ext_vector_type(8))) float v8f;

#define DEV __device__ __forceinline__

DEV void atomicMaxF(float* addr, float v) {
  if (v >= 0.0f) atomicMax((int*)addr, __float_as_int(v));
  else           atomicMin((unsigned int*)addr, (unsigned int)__float_as_int(v));
}

// ---------------- utility kernels ----------------
__global__ void fill_f32_k(float* __restrict__ p, float v, int n) {
  int i = blockIdx.x * blockDim.x + threadIdx.x;
  if (i < n) p[i] = v;
}

__global__ void pad_weights_k(const float* __restrict__ W, float* __restrict__ Wp,
                              int kin, int kout, int ncols) {
  int i = blockIdx.x * blockDim.x + threadIdx.x;
  if (i >= kout * ncols) return;
  int k = i / ncols, c = i - k * ncols;
  Wp[i] = (k < kin) ? W[k * ncols + c] : 0.0f;
}

// ---------------- batchnorm ----------------
// one block per feature column; deterministic LDS tree reduction
__global__ void bn_stats_k(const float* __restrict__ h, int n, int ld,
                           float* __restrict__ mean, float* __restrict__ rstd) {
  __shared__ float ssum[256], ssq[256];
  int f = blockIdx.x;
  float s = 0.f, q = 0.f;
  for (int i = threadIdx.x; i < n; i += 256) {
    float v = h[i * ld + f];
    s += v; q += v * v;
  }
  ssum[threadIdx.x] = s; ssq[threadIdx.x] = q;
  __syncthreads();
  for (int st = 128; st > 0; st >>= 1) {
    if (threadIdx.x < st) {
      ssum[threadIdx.x] += ssum[threadIdx.x + st];
      ssq[threadIdx.x]  += ssq[threadIdx.x + st];
    }
    __syncthreads();
  }
  if (threadIdx.x == 0) {
    float m = ssum[0] / (float)n;
    float var = ssq[0] / (float)n - m * m;
    mean[f] = m;
    rstd[f] = rsqrtf(var + 1e-5f);
  }
}

__global__ void bn_apply_k(const float* __restrict__ in, float* __restrict__ out,
                           const float* __restrict__ mean, const float* __restrict__ rstd,
                           const float* __restrict__ g, const float* __restrict__ b,
                           int n, int f, int ldin, int ldout, int relu) {
  int i = blockIdx.x * blockDim.x + threadIdx.x;
  if (i >= n * ldout) return;
  int row = i / ldout, c = i - row * ldout;
  float v = 0.0f;
  if (c < f) {
    v = (in[row * ldin + c] - mean[c]) * rstd[c] * g[c] + b[c];
    if (relu) v = fmaxf(v, 0.0f);
  }
  out[i] = v;
}

// ---------------- WMMA fp32 GEMM: out[M,Ncols] = A[M,K] @ W[K,Ncols] + bias ----
// one wave32 per 16x16 output tile; K multiple of 4; M multiple of 16; Ncols mult of 16
__global__ void wmma_gemm_f32_k(const float* __restrict__ A, const float* __restrict__ W,
                                const float* __restrict__ bias, float* __restrict__ out,
                                int K, int Ncols, int lda, int ldw, int ldo) {
  int lane = threadIdx.x;            // 0..31
  int tilesN = Ncols >> 4;
  int tile = blockIdx.x;
  int tm = tile / tilesN;
  int tn = tile - tm * tilesN;
  int half  = lane >> 4;             // K-half selector (ISA 16x4 A layout)
  int idx16 = lane & 15;
  int row = tm * 16 + idx16;         // A: M index
  int col = tn * 16 + idx16;         // B: N index
  v8f acc = {};
  for (int k0 = 0; k0 < K; k0 += 4) {
    int ka = k0 + half * 2;
    v2f a, b;
    a.x = A[row * lda + ka];
    a.y = A[row * lda + ka + 1];
    b.x = W[ka * ldw + col];
    b.y = W[(ka + 1) * ldw + col];
    acc = __builtin_amdgcn_wmma_f32_16x16x4_f32(false, a, false, b,
                                                (short)0, acc, false, false);
  }
  float bv = bias[col];
  #pragma unroll
  for (int r = 0; r < 8; ++r) {
    int orow = tm * 16 + r + half * 8;   // C/D layout: low lanes M=r, high lanes M=r+8
    out[orow * ldo + col] = acc[r] + bv;
  }
}

// ---------------- degree ----------------
__global__ void deg_k(const int* __restrict__ dst, float* __restrict__ deg, int E) {
  int i = blockIdx.x * blockDim.x + threadIdx.x;
  if (i < E) atomicAdd(&deg[dst[i]], 1.0f);
}

// ---------------- GATv2 edge kernels (wave32 per edge-head) ----------------
__global__ void edge_alpha_k(const float* __restrict__ xl, const float* __restrict__ xr,
                             const float* __restrict__ ea, const float* __restrict__ We,
                             const float* __restrict__ att,
                             const int* __restrict__ src, const int* __restrict__ dst,
                             float* __restrict__ alphaE, float* __restrict__ amax,
                             int E, int H, int C) {
  int g = blockIdx.x * blockDim.x + threadIdx.x;
  int w = g >> 5, lane = g & 31;
  if (w >= E * H) return;
  int e = w / H, h = w - e * H;
  int s = src[e], d = dst[e];
  float eav = ea[e];
  int HC = H * C;
  float part = 0.f;
  for (int c = lane; c < C; c += 32) {
    float v = xl[s * HC + h * C + c] + xr[d * HC + h * C + c] + eav * We[h * C + c];
    v = (v > 0.f) ? v : 0.2f * v;                // leaky_relu(0.2)
    part += v * att[h * C + c];
  }
  for (int o = 16; o; o >>= 1) part += __shfl_down(part, o, 32);
  if (lane == 0) {
    alphaE[w] = part;
    atomicMaxF(&amax[d * H + h], part);
  }
}

__global__ void edge_exp_k(float* __restrict__ alphaE, const float* __restrict__ amax,
                           float* __restrict__ denom, const int* __restrict__ dst,
                           int E, int H) {
  int i = blockIdx.x * blockDim.x + threadIdx.x;
  if (i >= E * H) return;
  int e = i / H, h = i - e * H;
  int d = dst[e];
  float ex = __expf(alphaE[i] - amax[d * H + h]);
  alphaE[i] = ex;
  atomicAdd(&denom[d * H + h], ex);
}

__global__ void edge_scatter_k(const float* __restrict__ xl, const float* __restrict__ alphaE,
                               const float* __restrict__ denom,
                               const int* __restrict__ src, const int* __restrict__ dst,
                               float* __restrict__ out, int E, int H, int C) {
  int g = blockIdx.x * blockDim.x + threadIdx.x;
  int w = g >> 5, lane = g & 31;
  if (w >= E * H) return;
  int e = w / H, h = w - e * H;
  int s = src[e], d = dst[e];
  int HC = H * C;
  float a = alphaE[w] / (denom[d * H + h] + 1e-16f);
  for (int c = lane; c < C; c += 32)
    atomicAdd(&out[d * HC + h * C + c], xl[s * HC + h * C + c] * a);
}

__global__ void node_scale_bias_k(float* __restrict__ out, const float* __restrict__ deg,
                                  const float* __restrict__ bias, int n, int HC) {
  int i = blockIdx.x * blockDim.x + threadIdx.x;
  if (i >= n * HC) return;
  int row = i / HC, c = i - row * HC;
  out[i] = out[i] / fmaxf(deg[row], 1.0f) + bias[c];
}

// ---------------- layer 3 (H=1, C=1) specialized ----------------
__global__ void dot2_128_k(const float* __restrict__ h, const float* __restrict__ Wl,
                           const float* __restrict__ Wr, const float* __restrict__ bl,
                           const float* __restrict__ br,
                           float* __restrict__ xl, float* __restrict__ xr, int n) {
  int g = blockIdx.x * blockDim.x + threadIdx.x;
  int node = g >> 5, lane = g & 31;
  if (node >= n) return;
  float pl = 0.f, pr = 0.f;
  for (int c = lane; c < 128; c += 32) {
    float hv = h[node * 128 + c];
    pl += hv * Wl[c];
    pr += hv * Wr[c];
  }
  for (int o = 16; o; o >>= 1) { pl += __shfl_down(pl, o, 32); pr += __shfl_down(pr, o, 32); }
  if (lane == 0) { xl[node] = pl + bl[0]; xr[node] = pr + br[0]; }
}

__global__ void edge3_alpha_k(const float* __restrict__ xl, const float* __restrict__ xr,
                              const float* __restrict__ ea, const float* __restrict__ We,
                              const float* __restrict__ att,
                              const int* __restrict__ src, const int* __restrict__ dst,
                              float* __restrict__ alphaE, float* __restrict__ amax, int E) {
  int i = blockIdx.x * blockDim.x + threadIdx.x;
  if (i >= E) return;
  int s = src[i], d = dst[i];
  float v = xl[s] + xr[d] + ea[i] * We[0];
  v = (v > 0.f) ? v : 0.2f * v;
  float a = v * att[0];
  alphaE[i] = a;
  atomicMaxF(&amax[d], a);
}

__global__ void edge3_scatter_k(const float* __restrict__ xl, const float* __restrict__ alphaE,
                                const float* __restrict__ denom,
                                const int* __restrict__ src, const int* __restrict__ dst,
                                float* __restrict__ out, int E) {
  int i = blockIdx.x * blockDim.x + threadIdx.x;
  if (i >= E) return;
  int s = src[i], d = dst[i];
  float a = alphaE[i] / (denom[d] + 1e-16f);
  atomicAdd(&out[d], xl[s] * a);
}

__global__ void final_k(const float* __restrict__ out3, const float* __restrict__ deg,
                        const float* __restrict__ bias, const float* __restrict__ y,
                        const int* __restrict__ mask, float* __restrict__ o, int n) {
  int i = blockIdx.x * blockDim.x + threadIdx.x;
  if (i >= n) return;
  float v = out3[i] / fmaxf(deg[i], 1.0f) + bias[0];
  float p = 1.0f / (1.0f + __expf(-v));
  bool m = (mask[i] != 0);
  o[i]     = m ? p    : 0.0f;
  o[n + i] = m ? y[i] : 0.0f;
}

// ---------------- launch ----------------
extern "C" void kernel_launch(void* const* d_in, const int* in_sizes, int n_in,
                              void* d_out, int out_size, void* d_ws, size_t ws_size,
                              hipStream_t stream) {
  const float* x     = (const float*)d_in[0];
  const int*   eidx  = (const int*)d_in[1];
  const float* ea    = (const float*)d_in[2];
  const float* y     = (const float*)d_in[3];
  const int*   mask  = (const int*)d_in[4];
  const float* bn0_g = (const float*)d_in[5];
  const float* bn0_b = (const float*)d_in[6];
  const float *Wl1 = (const float*)d_in[7],  *bl1 = (const float*)d_in[8];
  const float *Wr1 = (const float*)d_in[9],  *br1 = (const float*)d_in[10];
  const float *We1 = (const float*)d_in[11], *att1 = (const float*)d_in[12], *bias1 = (const float*)d_in[13];
  const float *bn1_g = (const float*)d_in[14], *bn1_b = (const float*)d_in[15];
  const float *Wl2 = (const float*)d_in[16], *bl2 = (const float*)d_in[17];
  const float *Wr2 = (const float*)d_in[18], *br2 = (const float*)d_in[19];
  const float *We2 = (const float*)d_in[20], *att2 = (const float*)d_in[21], *bias2 = (const float*)d_in[22];
  const float *bn2_g = (const float*)d_in[23], *bn2_b = (const float*)d_in[24];
  const float *Wl3 = (const float*)d_in[25], *bl3 = (const float*)d_in[26];
  const float *Wr3 = (const float*)d_in[27], *br3 = (const float*)d_in[28];
  const float *We3 = (const float*)d_in[29], *att3 = (const float*)d_in[30], *bias3 = (const float*)d_in[31];
  (void)n_in; (void)out_size; (void)ws_size;

  const int N = in_sizes[3];        // 50000
  const int E = in_sizes[1] / 2;    // 550000
  const int F = in_sizes[5];        // 13
  const int* src = eidx;
  const int* dst = eidx + E;

  // workspace carve-up (floats, 256B-aligned chunks)
  float* w = (float*)d_ws;
  size_t off = 0;
  auto alloc = [&](size_t count) { float* p = w + off; off += (count + 63) & ~(size_t)63; return p; };
  float* h0   = alloc((size_t)N * 16);
  float* bufA = alloc((size_t)N * 256);   // xl1 / xl2 / xl3
  float* bufB = alloc((size_t)N * 256);   // xr1 / xr2 / xr3
  float* bufC = alloc((size_t)N * 256);   // accum1 -> h1 ; out3
  float* bufI = alloc((size_t)N * 128);   // accum2 -> h2
  float* alE  = alloc((size_t)E * 2);
  float* amax = alloc((size_t)N * 2);
  float* den  = alloc((size_t)N * 2);
  float* deg  = alloc((size_t)N);
  float* mean = alloc(256);
  float* rstd = alloc(256);
  float* Wpl  = alloc(16 * 256);
  float* Wpr  = alloc(16 * 256);

  const int TB = 256;
  auto gs = [](long long n, int b) { return (unsigned)((n + b - 1) / b); };
  const float NEG_INF = -__builtin_huge_valf();
  const int tilesM = N / 16;   // N is a multiple of 16

  // ---- BN0 -> h0 (pad 13 -> 16 cols with zeros) ----
  bn_stats_k<<<F, 256, 0, stream>>>(x, N, F, mean, rstd);
  bn_apply_k<<<gs((long long)N * 16, TB), TB, 0, stream>>>(x, h0, mean, rstd, bn0_g, bn0_b, N, F, F, 16, 0);

  // ---- pad L1 weights K:13 -> 16 ----
  pad_weights_k<<<gs(16 * 256, TB), TB, 0, stream>>>(Wl1, Wpl, F, 16, 256);
  pad_weights_k<<<gs(16 * 256, TB), TB, 0, stream>>>(Wr1, Wpr, F, 16, 256);

  // ---- degree (shared by all layers) ----
  hipMemsetAsync(deg, 0, (size_t)N * sizeof(float), stream);
  deg_k<<<gs(E, TB), TB, 0, stream>>>(dst, deg, E);

  // ======== Layer 1 (H=2, C=128) ========
  wmma_gemm_f32_k<<<tilesM * (256 / 16), 32, 0, stream>>>(h0, Wpl, bl1, bufA, 16, 256, 16, 256, 256);
  wmma_gemm_f32_k<<<tilesM * (256 / 16), 32, 0, stream>>>(h0, Wpr, br1, bufB, 16, 256, 16, 256, 256);

  fill_f32_k<<<gs((long long)N * 2, TB), TB, 0, stream>>>(amax, NEG_INF, N * 2);
  hipMemsetAsync(den, 0, (size_t)N * 2 * sizeof(float), stream);
  hipMemsetAsync(bufC, 0, (size_t)N * 256 * sizeof(float), stream);
  long long thr1 = (long long)E * 2 * 32;
  edge_alpha_k<<<gs(thr1, TB), TB, 0, stream>>>(bufA, bufB, ea, We1, att1, src, dst, alE, amax, E, 2, 128);
  edge_exp_k<<<gs((long long)E * 2, TB), TB, 0, stream>>>(alE, amax, den, dst, E, 2);
  edge_scatter_k<<<gs(thr1, TB), TB, 0, stream>>>(bufA, alE, den, src, dst, bufC, E, 2, 128);
  node_scale_bias_k<<<gs((long long)N * 256, TB), TB, 0, stream>>>(bufC, deg, bias1, N, 256);

  // BN1 + ReLU in place
  bn_stats_k<<<256, 256, 0, stream>>>(bufC, N, 256, mean, rstd);
  bn_apply_k<<<gs((long long)N * 256, TB), TB, 0, stream>>>(bufC, bufC, mean, rstd, bn1_g, bn1_b, N, 256, 256, 256, 1);

  // ======== Layer 2 (H=1, C=128) ========
  wmma_gemm_f32_k<<<tilesM * (128 / 16), 32, 0, stream>>>(bufC, Wl2, bl2, bufA, 256, 128, 256, 128, 128);
  wmma_gemm_f32_k<<<tilesM * (128 / 16), 32, 0, stream>>>(bufC, Wr2, br2, bufB, 256, 128, 256, 128, 128);

  fill_f32_k<<<gs(N, TB), TB, 0, stream>>>(amax, NEG_INF, N);
  hipMemsetAsync(den, 0, (size_t)N * sizeof(float), stream);
  hipMemsetAsync(bufI, 0, (size_t)N * 128 * sizeof(float), stream);
  long long thr2 = (long long)E * 32;
  edge_alpha_k<<<gs(thr2, TB), TB, 0, stream>>>(bufA, bufB, ea, We2, att2, src, dst, alE, amax, E, 1, 128);
  edge_exp_k<<<gs(E, TB), TB, 0, stream>>>(alE, amax, den, dst, E, 1);
  edge_scatter_k<<<gs(thr2, TB), TB, 0, stream>>>(bufA, alE, den, src, dst, bufI, E, 1, 128);
  node_scale_bias_k<<<gs((long long)N * 128, TB), TB, 0, stream>>>(bufI, deg, bias2, N, 128);

  // BN2 + ReLU in place
  bn_stats_k<<<128, 256, 0, stream>>>(bufI, N, 128, mean, rstd);
  bn_apply_k<<<gs((long long)N * 128, TB), TB, 0, stream>>>(bufI, bufI, mean, rstd, bn2_g, bn2_b, N, 128, 128, 128, 1);

  // ======== Layer 3 (H=1, C=1) ========
  dot2_128_k<<<gs((long long)N * 32, TB), TB, 0, stream>>>(bufI, Wl3, Wr3, bl3, br3, bufA, bufB, N);
  fill_f32_k<<<gs(N, TB), TB, 0, stream>>>(amax, NEG_INF, N);
  hipMemsetAsync(den, 0, (size_t)N * sizeof(float), stream);
  hipMemsetAsync(bufC, 0, (size_t)N * sizeof(float), stream);
  edge3_alpha_k<<<gs(E, TB), TB, 0, stream>>>(bufA, bufB, ea, We3, att3, src, dst, alE, amax, E);
  edge_exp_k<<<gs(E, TB), TB, 0, stream>>>(alE, amax, den, dst, E, 1);
  edge3_scatter_k<<<gs(E, TB), TB, 0, stream>>>(bufA, alE, den, src, dst, bufC, E);

  final_k<<<gs(N, TB), TB, 0, stream>>>(bufC, deg, bias3, y, mask, (float*)d_out, N);
}